// FashionCaps_50294067036259
// MI455X (gfx1250) — compile-verified
//
#include <hip/hip_runtime.h>

typedef float v2f __attribute__((ext_vector_type(2)));
typedef float v8f __attribute__((ext_vector_type(8)));

constexpr int B_TOT = 64, I_TOT = 2048, K_TOT = 16, N_CAP = 32, D_CAP = 32;
constexpr int BBLK = 16;          // batch elements per workgroup
constexpr int IPW  = 16;          // input capsules per workgroup
constexpr int THREADS = 256;      // 8 wave32
constexpr int NBLK_B = B_TOT / BBLK;   // 4
constexpr int NBLK_I = I_TOT / IPW;    // 128
constexpr int TILE_ELEMS = BBLK * N_CAP * D_CAP;  // 16384
constexpr int WPAD = 20;          // W row pad: 16B-aligned rows + conflict-free ds reads

// ---- CDNA5 async global->LDS copy (ASYNCcnt path) ----
// Probe-derived signatures: src is payload-typed AS1 (non-const), dst AS3.
typedef int i32x4_t __attribute__((vector_size(16)));
typedef __attribute__((address_space(1))) i32x4_t GlobI4;
typedef __attribute__((address_space(3))) i32x4_t LdsI4;
typedef __attribute__((address_space(1))) int GlobI;
typedef __attribute__((address_space(3))) int LdsI;

__device__ __forceinline__ void async_ld_b128(const float* g, float* l) {
#if __has_builtin(__builtin_amdgcn_global_load_async_to_lds_b128)
  __builtin_amdgcn_global_load_async_to_lds_b128((GlobI4*)g, (LdsI4*)l, 0, 0);
#else
  unsigned lo = (unsigned)(uintptr_t)l;   // LDS offset = flat addr[31:0]
  asm volatile("global_load_async_to_lds_b128 %0, %1, off" :: "v"(lo), "v"(g) : "memory");
#endif
}
__device__ __forceinline__ void async_ld_b32(const float* g, float* l) {
#if __has_builtin(__builtin_amdgcn_global_load_async_to_lds_b32)
  __builtin_amdgcn_global_load_async_to_lds_b32((GlobI*)g, (LdsI*)l, 0, 0);
#else
  unsigned lo = (unsigned)(uintptr_t)l;
  asm volatile("global_load_async_to_lds_b32 %0, %1, off" :: "v"(lo), "v"(g) : "memory");
#endif
}
__device__ __forceinline__ void async_wait0() {
#if __has_builtin(__builtin_amdgcn_s_wait_asynccnt)
  __builtin_amdgcn_s_wait_asynccnt(0);
#else
  asm volatile("s_wait_asynccnt 0" ::: "memory");
#endif
}

// One routing iteration, fused: recompute u_hat (f32 WMMA) per input capsule,
// agreement update + logits + softmax(c) in LDS, accumulate s = sum_i c*u_hat
// into a deterministic per-block partial buffer. W/inputs double-buffered via
// async-to-LDS so iteration i+1's 64KB W tile streams in during i's compute.
__global__ __launch_bounds__(THREADS)
void caps_routing_pass(const float* __restrict__ inp,   // [B,I,K]
                       const float* __restrict__ Wt,    // [N,I,D,K]
                       const float* __restrict__ outp,  // [B,N,D] prev outputs
                       float* __restrict__ blog,        // [B,N,I] routing logits
                       float* __restrict__ part,        // [gridDim.x][BBLK*N*D]
                       int first)
{
  __shared__ __align__(16) float W_lds[2][N_CAP][D_CAP][WPAD];  // 2x80KB
  __shared__ float uh_lds[BBLK * N_CAP * 33];                    // d padded 32->33
  __shared__ float s_loc[TILE_ELEMS];
  __shared__ float inp_lds[2][BBLK][18];
  __shared__ float logit_lds[BBLK][N_CAP];
  __shared__ float c_lds[BBLK][N_CAP];

  const int tid  = threadIdx.x;
  const int wave = __builtin_amdgcn_readfirstlane(tid >> 5);  // uniform
  const int lane = tid & 31;
  const int lh   = lane >> 4;     // 0/1 : lane half (wave32 WMMA layout)
  const int l16  = lane & 15;

  const int bblk = blockIdx.x & (NBLK_B - 1);   // consecutive blocks share ichunk -> W L2 reuse
  const int ich  = blockIdx.x / NBLK_B;
  const int bg   = bblk * BBLK;
  const int i0   = ich * IPW;

  // issue the async stage for a given i into buffer `buf`
  auto stage = [&](int i, int buf) {
    const float* Wi = Wt + (size_t)i * (D_CAP * K_TOT);
    for (int idx = tid; idx < N_CAP * D_CAP * K_TOT / 4; idx += THREADS) {
      int n = idx >> 7, rem = idx & 127;        // 128 float4 per n (d,k contiguous)
      async_ld_b128(Wi + (size_t)n * (I_TOT * D_CAP * K_TOT) + rem * 4,
                    &W_lds[buf][n][rem >> 2][(rem & 3) * 4]);
    }
    // 256 threads == BBLK*K_TOT inputs
    int b = tid >> 4, k = tid & 15;
    async_ld_b32(inp + ((size_t)(bg + b) * I_TOT + i) * K_TOT + k,
                 &inp_lds[buf][b][k]);
  };

  for (int idx = tid; idx < TILE_ELEMS; idx += THREADS) s_loc[idx] = 0.f;

  stage(i0, 0);
  async_wait0();
  __syncthreads();

  for (int ii = 0; ii < IPW; ++ii) {
    const int i = i0 + ii;
    const int cur = ii & 1;

    if (ii + 1 < IPW) stage(i + 1, cur ^ 1);    // overlap next tile with compute

    // ---- u_hat[b,n,d] = sum_k inp[b,k] * W[n,i,d,k] via V_WMMA_F32_16X16X4_F32 ----
    // A (16x4 f32): lane = M (b), comp j -> K = 4*ks + 2*lh + j. Hoisted over n.
    v2f afrag[4];
    #pragma unroll
    for (int ks = 0; ks < 4; ++ks) {
      const float* p = &inp_lds[cur][l16][ks * 4 + 2 * lh];
      afrag[ks] = (v2f){p[0], p[1]};
    }
    // 64 output tiles (32 n x 2 d-tiles) over 8 waves; 4 chained k-step WMMAs each
    for (int t = wave; t < N_CAP * 2; t += THREADS / 32) {
      const int n = t >> 1, dt = t & 1;
      v8f acc = {0.f, 0.f, 0.f, 0.f, 0.f, 0.f, 0.f, 0.f};
      #pragma unroll
      for (int ks = 0; ks < 4; ++ks) {
        const float* q = &W_lds[cur][n][dt * 16 + l16][ks * 4 + 2 * lh];
        v2f bfrag = (v2f){q[0], q[1]};
        acc = __builtin_amdgcn_wmma_f32_16x16x4_f32(
                false, afrag[ks], false, bfrag, (short)0, acc, false, false);
      }
      #pragma unroll
      for (int v = 0; v < 8; ++v)   // D: VGPR v -> M = v + 8*lh, N = l16
        uh_lds[((v + 8 * lh) * N_CAP + n) * 33 + (dt * 16 + l16)] = acc[v];
    }
    __syncthreads();

    if (!first) {
      // agreement: logits[b,n] = blog[b,n,i] + sum_d out[b,n,d]*u_hat[b,n,d]
      // (outp is 256KB total -> L2-resident; float4 loads)
      for (int p = tid; p < BBLK * N_CAP; p += THREADS) {
        int b = p >> 5, n = p & 31;
        const float* uh = &uh_lds[(b * N_CAP + n) * 33];
        const float4* ob = (const float4*)(outp + ((size_t)(bg + b) * N_CAP + n) * D_CAP);
        float agr = 0.f;
        #pragma unroll
        for (int d4 = 0; d4 < D_CAP / 4; ++d4) {
          float4 o = ob[d4];
          agr = fmaf(o.x, uh[d4 * 4 + 0],
                fmaf(o.y, uh[d4 * 4 + 1],
                fmaf(o.z, uh[d4 * 4 + 2],
                fmaf(o.w, uh[d4 * 4 + 3], agr))));
        }
        size_t bi = ((size_t)(bg + b) * N_CAP + n) * I_TOT + i;
        float l = blog[bi] + agr;
        blog[bi] = l;
        logit_lds[b][n] = l;
      }
      __syncthreads();
      // softmax over n (stable), 16 threads each own one b
      if (tid < BBLK) {
        float m = -3.402823466e38f;
        #pragma unroll
        for (int n = 0; n < N_CAP; ++n) m = fmaxf(m, logit_lds[tid][n]);
        float sum = 0.f;
        #pragma unroll
        for (int n = 0; n < N_CAP; ++n) sum += __expf(logit_lds[tid][n] - m);
        float inv = 1.f / sum;
        #pragma unroll
        for (int n = 0; n < N_CAP; ++n)
          c_lds[tid][n] = __expf(logit_lds[tid][n] - m) * inv;
      }
      __syncthreads();
    }

    // s_loc += c * u_hat
    for (int idx = tid; idx < TILE_ELEMS; idx += THREADS) {
      int d = idx & 31, n = (idx >> 5) & 31, b = idx >> 10;
      float c = first ? (1.f / 32.f) : c_lds[b][n];
      s_loc[idx] = fmaf(c, uh_lds[(b * N_CAP + n) * 33 + d], s_loc[idx]);
    }

    // next tile must be fully landed in LDS before anyone reads it
    async_wait0();
    __syncthreads();
  }

  // deterministic partial (no float atomics): coalesced store
  float* my_part = part + (size_t)blockIdx.x * TILE_ELEMS;
  for (int idx = tid; idx < TILE_ELEMS; idx += THREADS) my_part[idx] = s_loc[idx];
}

// Sum the 128 i-chunk partials per (b,n,d) in a fixed order, then squash.
__global__ __launch_bounds__(32)
void caps_reduce_squash(const float* __restrict__ part, float* __restrict__ dst)
{
  const int pair = blockIdx.x;              // b*N_CAP + n
  const int d = threadIdx.x;                // 0..31
  const int b = pair >> 5;
  const int bblk = b >> 4;
  const int loc = ((b & (BBLK - 1)) * N_CAP + (pair & 31)) * D_CAP + d;
  float s = 0.f;
  for (int ic = 0; ic < NBLK_I; ++ic)
    s += part[(size_t)(ic * NBLK_B + bblk) * TILE_ELEMS + loc];
  float sq = s * s;
  #pragma unroll
  for (int off = 16; off > 0; off >>= 1) sq += __shfl_xor(sq, off, 32);
  float scale = sq / (1.f + sq) * rsqrtf(sq + 1e-7f);
  dst[(size_t)pair * D_CAP + d] = scale * s;
}

__global__ void caps_zero(float* __restrict__ p, int n) {
  int i = blockIdx.x * blockDim.x + threadIdx.x;
  if (i < n) p[i] = 0.f;
}

extern "C" void kernel_launch(void* const* d_in, const int* in_sizes, int n_in,
                              void* d_out, int out_size, void* d_ws, size_t ws_size,
                              hipStream_t stream) {
  const float* inp = (const float*)d_in[0];   // [64,2048,16]
  const float* Wt  = (const float*)d_in[1];   // [32,2048,32,16]
  float* ws   = (float*)d_ws;
  float* outp = ws;                                   //   65,536 f32
  float* blog = ws + 65536;                           // 4,194,304 f32
  float* part = ws + 65536 + (B_TOT * N_CAP * I_TOT); // 512*16384 f32 (32MB)
  float* dout = (float*)d_out;                        // [64,32,32]

  const int nblocks = NBLK_I * NBLK_B;                // 512
  caps_zero<<<(B_TOT * N_CAP * I_TOT + 255) / 256, 256, 0, stream>>>(blog, B_TOT * N_CAP * I_TOT);

  // r = 0 : c uniform
  caps_routing_pass<<<nblocks, THREADS, 0, stream>>>(inp, Wt, outp, blog, part, 1);
  caps_reduce_squash<<<B_TOT * N_CAP, 32, 0, stream>>>(part, outp);
  // r = 1 : logits = 0 + out0*u_hat
  caps_routing_pass<<<nblocks, THREADS, 0, stream>>>(inp, Wt, outp, blog, part, 0);
  caps_reduce_squash<<<B_TOT * N_CAP, 32, 0, stream>>>(part, outp);
  // r = 2 : logits = b1 + out1*u_hat ; final squash -> d_out
  caps_routing_pass<<<nblocks, THREADS, 0, stream>>>(inp, Wt, outp, blog, part, 0);
  caps_reduce_squash<<<B_TOT * N_CAP, 32, 0, stream>>>(part, dout);
}